// LieGroupEquivariantNeuralNetwork_2405181686614
// MI455X (gfx1250) — compile-verified
//
#include <hip/hip_runtime.h>

typedef __attribute__((ext_vector_type(2))) float v2f;
typedef __attribute__((ext_vector_type(8))) float v8f;

#define NPTS 96
#define NREP 4
#define MAXD 7
#define NCG  4
#define NCH  16
#define KDIM  (NPTS*NREP*MAXD)   // 2688 = j*(l,s)
#define KDIMP 2816               // padded to 44 tiles of 64 (4 slices x 11 tiles)
#define NDIM  (NREP*NCH*MAXD)    // 448  = (q,d,r)
#define NSLICE 4
#define KSLICE (KDIMP/NSLICE)    // 704

#if defined(__gfx1250__) && __has_builtin(__builtin_amdgcn_global_load_async_to_lds_b128) && __has_builtin(__builtin_amdgcn_s_wait_asynccnt)
#define USE_ASYNC_LDS 1
// Builtin signature (from clang diagnostic): (AS1 int4*, AS3 int4*, imm int, imm int)
typedef int v4i_b128 __attribute__((vector_size(16)));
typedef v4i_b128 __attribute__((address_space(1)))* g128_t;   // global src
typedef v4i_b128 __attribute__((address_space(3)))* l128_t;   // LDS dst
#else
#define USE_ASYNC_LDS 0
#endif

__device__ __forceinline__ void cmac(float& ar, float& ai, float br, float bi,
                                     float cr, float ci) {
    ar = fmaf(br, cr, fmaf(-bi, ci, ar));
    ai = fmaf(br, ci, fmaf(bi, cr, ai));
}

// cgf[q,r,s,t] = sum_g cg[g,3,s,3,t,q,r] * f[layer,q,g]   (complex)
__global__ void cgf_build(const float* __restrict__ cg, const float* __restrict__ f,
                          float* __restrict__ cgf, int layer) {
    int tid = blockIdx.x * blockDim.x + threadIdx.x;
    if (tid >= NREP*MAXD*MAXD*MAXD) return;
    int t = tid % 7, s = (tid/7) % 7, r = (tid/49) % 7, q = tid / 343;
    float ar = 0.f, ai = 0.f;
    for (int g = 0; g < NCG; ++g) {
        long ci = (((((((long)g*NREP + 3)*MAXD + s)*NREP + 3)*MAXD + t)*NREP + q)*MAXD + r);
        long fi = (((long)layer*NREP + q)*NCG + g) * 2;
        cmac(ar, ai, cg[ci*2], cg[ci*2+1], f[fi], f[fi+1]);
    }
    long o = (((long)q*MAXD + r)*MAXD + s)*MAXD + t;
    cgf[o*2] = ar; cgf[o*2+1] = ai;
}

// F[x,plane,i, j*28 + q*7+r] = base + sum_{s,t} cgf[q,r,s,t]*dX_s*dX_t
__global__ void filt_build(const float* __restrict__ X, const float* __restrict__ cgf,
                           float* __restrict__ F) {
    int tid = blockIdx.x * blockDim.x + threadIdx.x;
    if (tid >= 2*NPTS*NPTS*28) return;
    int r = tid % 7, q = (tid/7) % 4;
    int j = (tid/28) % NPTS, i = (tid/(28*NPTS)) % NPTS, x = tid/(28*NPTS*NPTS);
    const float* Xi = X + (long)(x*NPTS + i)*MAXD*2;
    const float* Xj = X + (long)(x*NPTS + j)*MAXD*2;
    float dxr[7], dxi[7];
    #pragma unroll
    for (int s = 0; s < 7; ++s) {
        dxr[s] = Xj[s*2]   - Xi[s*2];
        dxi[s] = Xj[s*2+1] - Xi[s*2+1];
    }
    float ar = 0.f, ai = 0.f;
    for (int s = 0; s < 7; ++s)
        for (int t = 0; t < 7; ++t) {
            float pr = dxr[s]*dxr[t] - dxi[s]*dxi[t];
            float pi = dxr[s]*dxi[t] + dxi[s]*dxr[t];
            long o = (((long)q*MAXD + r)*MAXD + s)*MAXD + t;
            cmac(ar, ai, cgf[o*2], cgf[o*2+1], pr, pi);
        }
    if (q == 3) { ar += dxr[r]; ai += dxi[r]; }
    int kidx = j*28 + q*7 + r;
    F[(((long)x*2 + 0)*NPTS + i)*KDIMP + kidx] = ar;
    F[(((long)x*2 + 1)*NPTS + i)*KDIMP + kidx] = ai;
}

// VW[x,j,q,d,g,m,t] = sum_c V[x,j,m,c,t] * W[layer,q,c,g,d]
__global__ void vw_build(const float* __restrict__ Vin, const float* __restrict__ W,
                         float* __restrict__ VW, int layer) {
    int tid = blockIdx.x * blockDim.x + threadIdx.x;
    if (tid >= 2*NPTS*NREP*NCH*NCG*NREP*MAXD) return;
    int t = tid % 7; int tmp = tid / 7;
    int m = tmp & 3;  tmp >>= 2;
    int g = tmp & 3;  tmp >>= 2;
    int d = tmp & 15; tmp >>= 4;
    int q = tmp & 3;  tmp >>= 2;
    int j = tmp % NPTS, x = tmp / NPTS;
    float ar = 0.f, ai = 0.f;
    for (int c = 0; c < NCH; ++c) {
        long vi = (((long)(x*NPTS + j)*NREP + m)*NCH + c)*MAXD + t;
        long wi = ((((long)layer*NREP + q)*NCH + c)*NCG + g)*NCH + d;
        cmac(ar, ai, Vin[vi*2], Vin[vi*2+1], W[wi*2], W[wi*2+1]);
    }
    long o = (((((long)(x*NPTS + j)*NREP + q)*NCH + d)*NCG + g)*NREP + m)*MAXD + t;
    VW[o*2] = ar; VW[o*2+1] = ai;
}

// Kmat[x,plane, j*28+(l,s), (q,d,r)] = sum_{g,m,t} cg[g,l,s,m,t,q,r]*VW[x,j,q,d,g,m,t]
__global__ void k_build(const float* __restrict__ cg, const float* __restrict__ VW,
                        float* __restrict__ Kmat) {
    int tid = blockIdx.x * blockDim.x + threadIdx.x;
    if (tid >= 2*NPTS*28*NDIM) return;
    int n  = tid % NDIM;
    int ls = (tid/NDIM) % 28;
    int j  = (tid/(NDIM*28)) % NPTS;
    int x  = tid/(NDIM*28*NPTS);
    int l = ls/7, s = ls%7;
    int q = n/112, d = (n/7) & 15, r = n % 7;
    float ar = 0.f, ai = 0.f;
    for (int g = 0; g < NCG; ++g)
        for (int m = 0; m < NREP; ++m) {
            long cbase = ((((long)g*NREP + l)*MAXD + s)*NREP + m)*196 + q*MAXD + r;
            long vbase = (((((long)(x*NPTS + j)*NREP + q)*NCH + d)*NCG + g)*NREP + m)*MAXD;
            for (int t = 0; t < MAXD; ++t) {
                long ci = cbase + (long)t*28;
                cmac(ar, ai, cg[ci*2], cg[ci*2+1], VW[(vbase+t)*2], VW[(vbase+t)*2+1]);
            }
        }
    int kidx = j*28 + ls;
    Kmat[(((long)x*2 + 0)*KDIMP + kidx)*NDIM + n] = ar;
    Kmat[(((long)x*2 + 1)*KDIMP + kidx)*NDIM + n] = ai;
}

// One-time zero fill of the K-padding rows (never rewritten afterwards).
__global__ void zero_pad_F(float* __restrict__ F) {
    int tid = blockIdx.x * blockDim.x + threadIdx.x;
    if (tid >= 2*2*NPTS*(KDIMP-KDIM)) return;
    int c = tid & 127, row = tid >> 7;               // 384 rows of 128 pad cols
    F[(long)row*KDIMP + KDIM + c] = 0.f;
}
__global__ void zero_pad_K(float* __restrict__ Kmat) {
    int tid = blockIdx.x * blockDim.x + threadIdx.x;
    if (tid >= 2*2*(KDIMP-KDIM)*NDIM) return;
    int n  = tid % NDIM;
    int kk = (tid/NDIM) & 127;
    int xp = tid/(NDIM*128);
    Kmat[((long)xp*KDIMP + KDIM + kk)*NDIM + n] = 0.f;
}

// Complex GEMM, K split over NSLICE slices per 16x16 tile for occupancy.
// Part[slice][x][i][n][z] partial results; reduced deterministically afterwards.
__global__ void __launch_bounds__(32)
cplx_gemm_wmma(const float* __restrict__ F, const float* __restrict__ Kmat,
               float* __restrict__ Part) {
    __shared__ __attribute__((aligned(16))) float As[2][16][68]; // [plane][i][k], 16B-aligned rows
    __shared__ __attribute__((aligned(16))) float Bs[2][64][16]; // [plane][k][n]
    int b = blockIdx.x;
    int slice = b & 3; int t2 = b >> 2;
    int x  = t2 / 168;
    int mt = (t2 / 28) % 6;
    int nt = t2 % 28;
    int i0 = mt*16, n0 = nt*16;
    int lane = threadIdx.x;
    int half = lane >> 4, lr = lane & 15;
    v8f cre = {}; v8f cim = {};
    const float* F0 = F + ((long)x*2 + 0)*NPTS*KDIMP;
    const float* F1 = F + ((long)x*2 + 1)*NPTS*KDIMP;
    const float* K0 = Kmat + ((long)x*2 + 0)*KDIMP*NDIM;
    const float* K1 = Kmat + ((long)x*2 + 1)*KDIMP*NDIM;
    int kbeg = slice * KSLICE, kend = kbeg + KSLICE;     // 11 tiles of 64
    for (int k0 = kbeg; k0 < kend; k0 += 64) {
        __syncthreads();   // 1 wave: s_wait_dscnt 0 + wave barrier; prior frag reads retired
#if USE_ASYNC_LDS
        #pragma unroll
        for (int e = 0; e < 1024; e += 128) {            // 16B per lane per op
            int fi = e + lane*4;
            int row = fi >> 6, c = fi & 63;              // A: 16 x 64
            __builtin_amdgcn_global_load_async_to_lds_b128(
                (g128_t)(F0 + (long)(i0+row)*KDIMP + k0 + c),
                (l128_t)&As[0][row][c], 0, 0);
            __builtin_amdgcn_global_load_async_to_lds_b128(
                (g128_t)(F1 + (long)(i0+row)*KDIMP + k0 + c),
                (l128_t)&As[1][row][c], 0, 0);
            int rb = fi >> 4, cb = fi & 15;              // B: 64 x 16
            __builtin_amdgcn_global_load_async_to_lds_b128(
                (g128_t)(K0 + (long)(k0+rb)*NDIM + n0 + cb),
                (l128_t)&Bs[0][rb][cb], 0, 0);
            __builtin_amdgcn_global_load_async_to_lds_b128(
                (g128_t)(K1 + (long)(k0+rb)*NDIM + n0 + cb),
                (l128_t)&Bs[1][rb][cb], 0, 0);
        }
        __builtin_amdgcn_s_wait_asynccnt(0);
#else
        for (int e = lane; e < 1024; e += 32) {
            int row = e >> 6, c = e & 63;
            As[0][row][c] = F0[(long)(i0 + row)*KDIMP + k0 + c];
            As[1][row][c] = F1[(long)(i0 + row)*KDIMP + k0 + c];
        }
        for (int e = lane; e < 1024; e += 32) {
            int row = e >> 4, c = e & 15;
            Bs[0][row][c] = K0[(long)(k0 + row)*NDIM + n0 + c];
            Bs[1][row][c] = K1[(long)(k0 + row)*NDIM + n0 + c];
        }
#endif
        __syncthreads();
        #pragma unroll
        for (int kk = 0; kk < 64; kk += 4) {
            int ac = kk + 2*half;   // A: K = v + 2*half ; B: K = v + 2*half
            v2f ar, ai, br, bi, bin;
            ar.x = As[0][lr][ac];  ar.y = As[0][lr][ac+1];
            ai.x = As[1][lr][ac];  ai.y = As[1][lr][ac+1];
            br.x = Bs[0][ac][lr];  br.y = Bs[0][ac+1][lr];
            bi.x = Bs[1][ac][lr];  bi.y = Bs[1][ac+1][lr];
            bin = -bi;
            // Re += Ar*Br - Ai*Bi ; Im += Ar*Bi + Ai*Br
            cre = __builtin_amdgcn_wmma_f32_16x16x4_f32(false, ar, false, br,  (short)0, cre, false, false);
            cre = __builtin_amdgcn_wmma_f32_16x16x4_f32(false, ai, false, bin, (short)0, cre, false, false);
            cim = __builtin_amdgcn_wmma_f32_16x16x4_f32(false, ar, false, bi,  (short)0, cim, false, false);
            cim = __builtin_amdgcn_wmma_f32_16x16x4_f32(false, ai, false, br,  (short)0, cim, false, false);
        }
    }
    #pragma unroll
    for (int v = 0; v < 8; ++v) {      // D: M = v + 8*half, N = lr
        int i = i0 + v + 8*half;
        int n = n0 + lr;
        long o = (((long)(slice*2 + x)*NPTS + i)*NDIM + n) * 2;
        Part[o]   = cre[v];
        Part[o+1] = cim[v];
    }
}

// Deterministic fixed-order reduction of the 4 K-slices.
__global__ void reduce_part(const float* __restrict__ Part, float* __restrict__ Out) {
    int tid = blockIdx.x * blockDim.x + threadIdx.x;
    if (tid >= 2*NPTS*NDIM*2) return;
    const long st = (long)2*NPTS*NDIM*2;
    Out[tid] = ((Part[tid] + Part[tid + st]) + Part[tid + 2*st]) + Part[tid + 3*st];
}

extern "C" void kernel_launch(void* const* d_in, const int* in_sizes, int n_in,
                              void* d_out, int out_size, void* d_ws, size_t ws_size,
                              hipStream_t stream) {
    (void)in_sizes; (void)n_in; (void)out_size; (void)ws_size;
    const float* X  = (const float*)d_in[0];
    // d_in[1] = l (int scalar) -- fixed to 3 in this problem instance
    const float* V  = (const float*)d_in[2];
    const float* cg = (const float*)d_in[3];
    const float* f  = (const float*)d_in[4];
    const float* W  = (const float*)d_in[5];
    // d_in[6] = C4 complex contractor -- complex arithmetic is hardcoded

    float* ws = (float*)d_ws;
    size_t off = 0;
    float* Fmat = ws + off; off += (size_t)2*2*NPTS*KDIMP;                      // 1,081,344
    float* Kmat = ws + off; off += (size_t)2*2*KDIMP*NDIM;                      // 5,046,272
    float* VW   = ws + off; off += (size_t)2*NPTS*NREP*NCH*NCG*NREP*MAXD*2;     // 2,752,512
    float* cgf  = ws + off; off += (size_t)NREP*MAXD*MAXD*MAXD*2;               // 2,744
    float* Vb0  = ws + off; off += (size_t)2*NPTS*NDIM*2;                       // 172,032
    float* Vb1  = ws + off; off += (size_t)2*NPTS*NDIM*2;                       // 172,032
    float* Part = ws + off; off += (size_t)NSLICE*2*NPTS*NDIM*2;                // 688,128

    // one-time zero fill of K-padding (never rewritten)
    zero_pad_F<<<(2*2*NPTS*(KDIMP-KDIM))/256, 256, 0, stream>>>(Fmat);
    zero_pad_K<<<(2*2*(KDIMP-KDIM)*NDIM)/256, 256, 0, stream>>>(Kmat);

    for (int k = 0; k < 3; ++k) {
        const float* Vin = (k == 0) ? V : (k == 1 ? Vb0 : Vb1);
        float* Vout = (k == 2) ? (float*)d_out : (k == 0 ? Vb0 : Vb1);
        cgf_build <<<(NREP*MAXD*MAXD*MAXD + 255)/256, 256, 0, stream>>>(cg, f, cgf, k);
        filt_build<<<(2*NPTS*NPTS*28)/256,                256, 0, stream>>>(X, cgf, Fmat);
        vw_build  <<<(2*NPTS*NREP*NCH*NCG*NREP*MAXD)/256, 256, 0, stream>>>(Vin, W, VW, k);
        k_build   <<<(2*NPTS*28*NDIM)/256,                256, 0, stream>>>(cg, VW, Kmat);
        cplx_gemm_wmma<<<2*6*28*NSLICE, 32, 0, stream>>>(Fmat, Kmat, Part);
        reduce_part<<<(2*NPTS*NDIM*2)/256, 256, 0, stream>>>(Part, Vout);
    }
}